// HeatEquationGNN_85306640433889
// MI455X (gfx1250) — compile-verified
//
#include <hip/hip_runtime.h>

// ---- CDNA5 vector types ----
typedef __attribute__((ext_vector_type(4))) float     fx4;
typedef __attribute__((ext_vector_type(2))) float     v2f;
typedef __attribute__((ext_vector_type(8))) float     v8f;
typedef __attribute__((ext_vector_type(2))) long long lx2;

// ------------------------------------------------------------------
// Kernel 1: zero the per-node accumulator (graph-replay safe init).
// Vectorized b128 stores; scalar tail.
// ------------------------------------------------------------------
__global__ void hegnn_zero_f32(float* __restrict__ p, int n) {
    int i  = blockIdx.x * blockDim.x + threadIdx.x;
    int i4 = i * 4;
    if (i4 + 3 < n) {
        fx4 z = {0.f, 0.f, 0.f, 0.f};
        ((fx4*)p)[i] = z;
    } else {
        for (int k = i4; k < n; ++k) p[k] = 0.0f;
    }
}

// ------------------------------------------------------------------
// Kernel 2: per-edge message + scatter-add at target nodes.
// 4 edges per thread -> 3x b128 coalesced attr loads + 2x b128 index
// loads, all non-temporal (streaming 320 MB; keep L2 for the 4 MB
// accumulator).  msg_e = attr[e]·w_msg + b ;  aggr[tgt[e]] += msg_e
// via global_atomic_add_f32 (no return -> STOREcnt path).
// ------------------------------------------------------------------
__global__ void hegnn_edge_scatter(const float*     __restrict__ attr,
                                   const long long* __restrict__ tgt,
                                   const float*     __restrict__ w_msg,
                                   const float*     __restrict__ b_msg,
                                   float*           __restrict__ aggr,
                                   long long nEdges) {
    long long t  = (long long)blockIdx.x * blockDim.x + threadIdx.x;
    long long e0 = t * 4;
    if (e0 >= nEdges) return;

    const float w0 = w_msg[0], w1 = w_msg[1], w2 = w_msg[2], bm = b_msg[0];

    if (e0 + 3 < nEdges) {
        const fx4* a4 = (const fx4*)attr;       // 48B per thread, 16B aligned
        fx4 a = __builtin_nontemporal_load(&a4[3 * t + 0]);
        fx4 b = __builtin_nontemporal_load(&a4[3 * t + 1]);
        fx4 c = __builtin_nontemporal_load(&a4[3 * t + 2]);
        const lx2* t2 = (const lx2*)tgt;        // 32B per thread, 16B aligned
        lx2 i01 = __builtin_nontemporal_load(&t2[2 * t + 0]);
        lx2 i23 = __builtin_nontemporal_load(&t2[2 * t + 1]);

        float m0 = fmaf(a.x, w0, fmaf(a.y, w1, fmaf(a.z, w2, bm)));
        float m1 = fmaf(a.w, w0, fmaf(b.x, w1, fmaf(b.y, w2, bm)));
        float m2 = fmaf(b.z, w0, fmaf(b.w, w1, fmaf(c.x, w2, bm)));
        float m3 = fmaf(c.y, w0, fmaf(c.z, w1, fmaf(c.w, w2, bm)));

        unsafeAtomicAdd(&aggr[(int)i01.x], m0);
        unsafeAtomicAdd(&aggr[(int)i01.y], m1);
        unsafeAtomicAdd(&aggr[(int)i23.x], m2);
        unsafeAtomicAdd(&aggr[(int)i23.y], m3);
    } else {
        for (long long e = e0; e < nEdges; ++e) {
            float m = fmaf(attr[3 * e + 0], w0,
                      fmaf(attr[3 * e + 1], w1,
                      fmaf(attr[3 * e + 2], w2, bm)));
            unsafeAtomicAdd(&aggr[(int)tgt[e]], m);
        }
    }
}

// ------------------------------------------------------------------
// Kernel 3: node update via V_WMMA_F32_16X16X4_F32.
// Each wave handles 16 nodes: A = x[base:base+16, 0:4] (16x4, f32),
// B = w_upd[0:4] broadcast into all 16 columns (4x16), C = 0.
// D[m][*] = dot(x[base+m], w_upd[0:4]) exactly (f32 WMMA).
// Lanes 0 and 16 fold in the remaining scalar terms for 8 nodes each:
//   out[n] = x[n,0] + D[n] + aggr[n]*x[n,3]*w_upd[4] + b_upd
// Fast path (full tile): 8x b128 x-loads + 2x b128 aggr loads +
// 2x b128 packed stores, no per-node branches.
//
// A layout (32-bit 16x4): lanes 0-15 hold K=0,1 in v[0],v[1];
// lanes 16-31 hold K=2,3.  B mirrors with N across lanes.
// D layout: VGPR r, lane L -> M = r + (L>=16 ? 8 : 0), N = L&15.
// ------------------------------------------------------------------
__global__ void hegnn_node_update(const float* __restrict__ x,
                                  const float* __restrict__ aggr,
                                  const float* __restrict__ w_upd,
                                  const float* __restrict__ b_upd,
                                  float*       __restrict__ out,
                                  int nNodes) {
    const int wave = (int)((blockIdx.x * blockDim.x + threadIdx.x) >> 5);
    const int lane = threadIdx.x & 31;
    const int base = wave * 16;
    if (base >= nNodes) return;              // uniform per wave

    const int  m  = lane & 15;
    const bool hi = lane >= 16;
    int node = base + m;
    if (node >= nNodes) node = nNodes - 1;   // clamp (pad rows unused)

    const fx4 xv = ((const fx4*)x)[node];    // x is [N,4], 16B rows

    const float wu0 = w_upd[0], wu1 = w_upd[1], wu2 = w_upd[2],
                wu3 = w_upd[3], wu4 = w_upd[4];
    const float bu  = b_upd[0];

    v2f a, b;
    a[0] = hi ? xv.z : xv.x;                 // K = hi ? 2 : 0
    a[1] = hi ? xv.w : xv.y;                 // K = hi ? 3 : 1
    b[0] = hi ? wu2 : wu0;                   // same K rows, broadcast over N
    b[1] = hi ? wu3 : wu1;

    v8f c = {};
    // 8 args: (neg_a, A, neg_b, B, c_mod, C, reuse_a, reuse_b)
    c = __builtin_amdgcn_wmma_f32_16x16x4_f32(
            false, a, false, b, (short)0, c, false, false);

    if (m == 0) {                            // lanes 0 and 16 write results
        const int nb = base + (hi ? 8 : 0);  // nb % 8 == 0
        if (nb + 8 <= nNodes) {
            // ---- full-tile fast path: no per-node guards ----
            const fx4* x4 = (const fx4*)x;
            fx4 xr0 = x4[nb + 0], xr1 = x4[nb + 1], xr2 = x4[nb + 2], xr3 = x4[nb + 3];
            fx4 xr4 = x4[nb + 4], xr5 = x4[nb + 5], xr6 = x4[nb + 6], xr7 = x4[nb + 7];
            const fx4* ag4 = (const fx4*)(aggr + nb);     // 16B aligned
            fx4 ag0 = ag4[0], ag1 = ag4[1];

            fx4 o0, o1;
            o0.x = xr0.x + c[0] + ag0.x * xr0.w * wu4 + bu;
            o0.y = xr1.x + c[1] + ag0.y * xr1.w * wu4 + bu;
            o0.z = xr2.x + c[2] + ag0.z * xr2.w * wu4 + bu;
            o0.w = xr3.x + c[3] + ag0.w * xr3.w * wu4 + bu;
            o1.x = xr4.x + c[4] + ag1.x * xr4.w * wu4 + bu;
            o1.y = xr5.x + c[5] + ag1.y * xr5.w * wu4 + bu;
            o1.z = xr6.x + c[6] + ag1.z * xr6.w * wu4 + bu;
            o1.w = xr7.x + c[7] + ag1.w * xr7.w * wu4 + bu;

            fx4* o4 = (fx4*)(out + nb);                   // 32B aligned
            o4[0] = o0;
            o4[1] = o1;
        } else {
            // ---- guarded tail (last partial tile only) ----
#pragma unroll
            for (int r = 0; r < 8; ++r) {
                const int n = nb + r;
                if (n < nNodes) {
                    const fx4 xr = ((const fx4*)x)[n];
                    out[n] = xr.x + c[r] + aggr[n] * xr.w * wu4 + bu;
                }
            }
        }
    }
}

// ------------------------------------------------------------------
// Launch
// ------------------------------------------------------------------
extern "C" void kernel_launch(void* const* d_in, const int* in_sizes, int n_in,
                              void* d_out, int out_size, void* d_ws, size_t ws_size,
                              hipStream_t stream) {
    const float*     x     = (const float*)d_in[0];                 // [N,4]
    const long long* eidx  = (const long long*)d_in[1];             // [2,E] i64
    const float*     attr  = (const float*)d_in[2];                 // [E,3]
    const float*     w_msg = (const float*)d_in[3];                 // [3]
    const float*     b_msg = (const float*)d_in[4];                 // [1]
    const float*     w_upd = (const float*)d_in[5];                 // [5]
    const float*     b_upd = (const float*)d_in[6];                 // [1]

    const int       nNodes = in_sizes[0] / 4;
    const long long nEdges = (long long)in_sizes[2] / 3;
    const long long* tgt   = eidx + nEdges;                         // edge_index[1]

    float* aggr = (float*)d_ws;          // nNodes floats (4 MB), L2-resident
    float* out  = (float*)d_out;         // nNodes floats

    // 1) zero accumulator (b128 fills)
    {
        int threads = 256;
        int nVec    = (nNodes + 3) / 4;
        int blocks  = (nVec + threads - 1) / threads;
        hegnn_zero_f32<<<blocks, threads, 0, stream>>>(aggr, nNodes);
    }
    // 2) edge messages + scatter-add (4 edges / thread)
    {
        long long nT = (nEdges + 3) / 4;
        int threads  = 256;
        int blocks   = (int)((nT + threads - 1) / threads);
        hegnn_edge_scatter<<<blocks, threads, 0, stream>>>(
            attr, tgt, w_msg, b_msg, aggr, nEdges);
    }
    // 3) node update with WMMA (16 nodes / wave, 8 waves / block)
    {
        int waves   = (nNodes + 15) / 16;
        int threads = 256;                 // 8 wave32s
        int blocks  = (waves + 7) / 8;
        hegnn_node_update<<<blocks, threads, 0, stream>>>(
            x, aggr, w_upd, b_upd, out, nNodes);
    }
}